// OrdinalWassersteinLoss_80504866996362
// MI455X (gfx1250) — compile-verified
//
#include <hip/hip_runtime.h>
#include <hip/hip_bf16.h>

// Ordinal Wasserstein loss, MI455X (gfx1250, wave32).
//
// Roofline: 96 MiB preds + 32 MiB targets at 23.3 TB/s => ~6 us floor; fully
// memory-bound (VALU ~0.3 GFLOP total). Working set (128 MiB) fits the 192 MB
// L2, so default (RT) temporal hints are used to let replays hit L2.
//
// Data path: 8 rows/thread, all-b128 coalesced loads; compute BOTH the
// softmax-CDF loss and the raw-CDF loss plus the global "any pred outside
// [0,1]" flag in one pass (the traced jnp.where branch).
//
// Reduction path (CDNA5-specific): wave-level sums are reduced with
// V_WMMA_F32_16X16X4_F32 using an all-ones B matrix. A constant B is
// layout-invariant, and the A (16x4 f32) and C/D (16x16 f32) VGPR layouts are
// documented, so D[m][*] = accX[m] + accX[m+16]; summing D's 8 VGPRs plus one
// shfl_xor(16) yields the full 32-lane sum on the XDL unit, replacing the
// shuffle chain. Cross-wave reduce via LDS, then f64 global atomics.

typedef float f32x4 __attribute__((ext_vector_type(4)));
typedef int   i32x4 __attribute__((ext_vector_type(4)));
typedef float v2f   __attribute__((ext_vector_type(2)));
typedef float v8f   __attribute__((ext_vector_type(8)));

#define BLOCK 256
#define RPT   8   // rows per thread
#define WAVES (BLOCK / 32)

__global__ void owl_init(double* __restrict__ sums, unsigned* __restrict__ flag) {
    sums[0] = 0.0;   // softmax-branch loss sum
    sums[1] = 0.0;   // raw-branch loss sum
    *flag   = 0u;    // 1 => some pred outside [0,1] => softmax branch taken
}

__device__ __forceinline__ void row_loss(float p0, float p1, float p2, int t,
                                         float& accS, float& accR, unsigned& outside) {
    const float i0 = (t == 0) ? 1.0f : 0.0f;   // target CDF at class 0
    const float i1 = (t <= 1) ? 1.0f : 0.0f;   // target CDF at class 1
    // --- softmax branch ---
    const float m  = fmaxf(p0, fmaxf(p1, p2));
    const float e0 = __expf(p0 - m);
    const float e1 = __expf(p1 - m);
    const float e2 = __expf(p2 - m);
    const float inv = 1.0f / (e0 + e1 + e2);
    const float q0 = e0 * inv;
    const float c1 = q0 + e1 * inv;
    const float c2 = c1 + e2 * inv;
    accS += fabsf(q0 - i0) + fabsf(c1 - i1) + fabsf(c2 - 1.0f);
    // --- raw branch (no softmax) ---
    const float r1 = p0 + p1;
    const float r2 = r1 + p2;
    accR += fabsf(p0 - i0) + fabsf(r1 - i1) + fabsf(r2 - 1.0f);
    // --- global predicate: any element outside [0,1]? ---
    outside |= (unsigned)((p0 < 0.0f) | (p0 > 1.0f) |
                          (p1 < 0.0f) | (p1 > 1.0f) |
                          (p2 < 0.0f) | (p2 > 1.0f));
}

// Sum a per-lane f32 value across the full wave32 using the XDL unit.
// A: VGPR0 = x, VGPR1 = 0  =>  A[m][:] = (x[m], 0, x[m+16], 0)
// B: all-ones (layout-invariant constant matrix)
// D[m][n] = x[m] + x[m+16]  (replicated over n)
// C/D layout: VGPR r holds M=r (lanes 0-15) and M=r+8 (lanes 16-31).
// Per-lane sum of D's 8 VGPRs covers half the M range; shfl_xor(16) completes.
__device__ __forceinline__ float wave_sum_wmma(float x) {
    v2f a = {x, 0.0f};
    v2f b = {1.0f, 1.0f};
    v8f c = {0.0f, 0.0f, 0.0f, 0.0f, 0.0f, 0.0f, 0.0f, 0.0f};
    v8f d = __builtin_amdgcn_wmma_f32_16x16x4_f32(
        /*neg_a=*/false, a, /*neg_b=*/false, b,
        /*c_mod=*/(short)0, c, /*reuse_a=*/false, /*reuse_b=*/false);
    float s = ((d[0] + d[1]) + (d[2] + d[3])) + ((d[4] + d[5]) + (d[6] + d[7]));
    s += __shfl_xor(s, 16, 32);
    return s;   // full 32-lane sum, in every lane
}

__global__ __launch_bounds__(BLOCK)
void owl_main(const float* __restrict__ preds, const int* __restrict__ targets,
              double* __restrict__ sums, unsigned* __restrict__ flag, int batch) {
    const long long tid = (long long)blockIdx.x * BLOCK + threadIdx.x;
    const long long r0  = tid * RPT;

    float accS = 0.0f, accR = 0.0f;
    unsigned outside = 0u;

    if (r0 + RPT <= (long long)batch) {
        // 8 rows * 3 floats = 24 floats = 6 x b128 (96B group, 16B aligned)
        const f32x4* pv = (const f32x4*)(preds + r0 * 3);
        f32x4 v[6];
        #pragma unroll
        for (int i = 0; i < 6; ++i) v[i] = pv[i];
        // 8 int32 targets = 2 x b128
        const i32x4* tv = (const i32x4*)(targets + r0);
        i32x4 t0 = tv[0];
        i32x4 t1 = tv[1];
        #pragma unroll
        for (int i = 0; i < RPT; ++i) {
            const int b = 3 * i;
            const float p0 = v[(b    ) >> 2][(b    ) & 3];
            const float p1 = v[(b + 1) >> 2][(b + 1) & 3];
            const float p2 = v[(b + 2) >> 2][(b + 2) & 3];
            const int   t  = (i < 4) ? t0[i] : t1[i - 4];
            row_loss(p0, p1, p2, t, accS, accR, outside);
        }
    } else {
        // scalar tail (only the last partial block, if any)
        const long long rend = ((long long)batch < r0 + RPT) ? (long long)batch : (r0 + RPT);
        for (long long r = r0; r < rend; ++r) {
            const float p0 = preds[r * 3 + 0];
            const float p1 = preds[r * 3 + 1];
            const float p2 = preds[r * 3 + 2];
            row_loss(p0, p1, p2, targets[r], accS, accR, outside);
        }
    }

    // ---- wave32 reduction on the XDL unit (EXEC is all-ones here: control
    // flow has reconverged, every launched thread executes this path) ----
    const float sS = wave_sum_wmma(accS);
    const float sR = wave_sum_wmma(accR);
    const unsigned fl = (__ballot(outside != 0u) != 0ull) ? 1u : 0u;

    // ---- cross-wave reduction via LDS ----
    __shared__ double   ls[WAVES];
    __shared__ double   lr[WAVES];
    __shared__ unsigned lf[WAVES];
    const int wave = threadIdx.x >> 5;
    const int lane = threadIdx.x & 31;
    if (lane == 0) { ls[wave] = (double)sS; lr[wave] = (double)sR; lf[wave] = fl; }
    __syncthreads();

    if (threadIdx.x == 0) {
        double ts = 0.0, tr = 0.0; unsigned tf = 0u;
        #pragma unroll
        for (int i = 0; i < WAVES; ++i) { ts += ls[i]; tr += lr[i]; tf |= lf[i]; }
        atomicAdd(&sums[0], ts);          // global_atomic_add_f64
        atomicAdd(&sums[1], tr);
        if (tf) atomicOr(flag, 1u);
    }
}

__global__ void owl_fin(const double* __restrict__ sums, const unsigned* __restrict__ flag,
                        float* __restrict__ out, int batch) {
    const double sum  = (*flag != 0u) ? sums[0] : sums[1];
    out[0] = (float)(sum / (double)batch);
}

extern "C" void kernel_launch(void* const* d_in, const int* in_sizes, int n_in,
                              void* d_out, int out_size, void* d_ws, size_t ws_size,
                              hipStream_t stream) {
    const float* preds   = (const float*)d_in[0];
    const int*   targets = (const int*)d_in[1];   // harness contract: integer -> const int*
    const int    batch   = in_sizes[1];           // targets element count == BATCH

    double*   sums = (double*)d_ws;
    unsigned* flag = (unsigned*)((char*)d_ws + 2 * sizeof(double));
    float*    out  = (float*)d_out;

    owl_init<<<1, 1, 0, stream>>>(sums, flag);

    const long long rows_per_block = (long long)BLOCK * RPT;    // 2048
    const int nblocks = (int)(((long long)batch + rows_per_block - 1) / rows_per_block);
    owl_main<<<nblocks, BLOCK, 0, stream>>>(preds, targets, sums, flag, batch);

    owl_fin<<<1, 1, 0, stream>>>(sums, flag, out, batch);
}